// CustomAggregationLayer_35845797052840
// MI455X (gfx1250) — compile-verified
//
#include <hip/hip_runtime.h>

typedef __attribute__((ext_vector_type(2))) float v2f;
typedef __attribute__((ext_vector_type(4))) float v4f;
typedef __attribute__((ext_vector_type(8))) float v8f;

#define N_NODES  50000
#define DEG      32
#define D_FEAT   128
#define D_IN     256   // 2*D_FEAT
#define D_OUT    256
#define M_TILE   32
#define LDS_STR  260   // 256 + 4 pad -> conflict-free ds_load_b64 A-fragments

// ---------------------------------------------------------------------------
// Transpose W[k][n] -> Wt[n][k] so B fragments (two consecutive K rows at one
// column) become a single contiguous global_load_b64. 256KB once, L2-resident.
// ---------------------------------------------------------------------------
__global__ __launch_bounds__(256) void transpose_w(const float* __restrict__ W,
                                                   float* __restrict__ Wt)
{
    const int n = blockIdx.x;          // output row (= column of W)
    const int k = threadIdx.x;
    Wt[n * D_IN + k] = W[k * D_OUT + n];
}

// ---------------------------------------------------------------------------
// Fused: build X = [self | mean(neigh)] tile in LDS, then X @ W with f32 WMMA.
// TR: weights are pre-transposed (Wt[n][k]) in d_ws.
// ---------------------------------------------------------------------------
template <bool TR>
__global__ __launch_bounds__(256) void sage_fused_wmma(
    const float* __restrict__ feat,
    const long long* __restrict__ edges,
    const float* __restrict__ Wmat,     // TR ? Wt[n][k] : W[k][n]
    float* __restrict__ out)
{
    __shared__ float Xs[M_TILE][LDS_STR];

    const int tid  = threadIdx.x;
    const int lane = tid & 31;
    const int wave = tid >> 5;
    const int base = blockIdx.x * M_TILE;

    const bool full_tile = (base + M_TILE) <= N_NODES;

    // ---------- Phase 1a: self features -> Xs[:, 0:128] ----------
    if (full_tile) {
        // CDNA5 async global->LDS copy (ASYNCcnt), overlaps with phase 1b gather.
        #pragma unroll
        for (int i = tid; i < M_TILE * (D_FEAT / 4); i += 256) {
            const int nl = i >> 5;                 // local node
            const int c4 = i & 31;                 // float4 index within row
            const unsigned lds_off = (unsigned)((nl * LDS_STR + c4 * 4) * 4);
            const unsigned goff    = (unsigned)((base + nl) * (D_FEAT * 4) + c4 * 16);
            asm volatile("global_load_async_to_lds_b128 %0, %1, %2"
                         :: "v"(lds_off), "v"(goff), "s"(feat) : "memory");
        }
    } else {
        #pragma unroll
        for (int i = tid; i < M_TILE * (D_FEAT / 4); i += 256) {
            const int nl   = i >> 5;
            const int c4   = i & 31;
            const int node = base + nl;
            v4f v = {0.f, 0.f, 0.f, 0.f};
            if (node < N_NODES)
                v = *(const v4f*)&feat[(long long)node * D_FEAT + c4 * 4];
            *(v4f*)&Xs[nl][c4 * 4] = v;
        }
    }

    // ---------- Phase 1b: neighbor mean -> Xs[:, 128:256] ----------
    // 8 waves x 4 nodes; lanes cover the 128 feature columns (4 each).
    #pragma unroll
    for (int j = 0; j < M_TILE / 8; ++j) {
        const int nl   = wave * (M_TILE / 8) + j;
        const int node = base + nl;
        v4f acc = {0.f, 0.f, 0.f, 0.f};
        if (node < N_NODES) {
            const long long* e = edges + (long long)node * DEG;  // wave-uniform -> s_loads
            #pragma unroll 4
            for (int n = 0; n < DEG; ++n) {
                const long long nbr = e[n];
                acc += *(const v4f*)&feat[nbr * D_FEAT + lane * 4];
            }
            acc *= (1.0f / (float)DEG);
        }
        *(v4f*)&Xs[nl][D_FEAT + lane * 4] = acc;
    }

    if (full_tile)
        asm volatile("s_wait_asynccnt 0x0" ::: "memory");
    __syncthreads();

    // ---------- Phase 2: [32 x 256] @ [256 x 256] via v_wmma_f32_16x16x4_f32 ---
    // wave -> (M-tile = wave&1, N-tiles = (wave>>1)*4 .. +3)
    const int m0    = (wave & 1) * 16;
    const int nbase = (wave >> 1) * 64;

    // A fragment (16x4 f32): lanes 0-15 row M=lane, VGPR0/1 = K 0/1;
    //                        lanes 16-31 same rows,  VGPR0/1 = K 2/3.
    const int row   = lane & 15;
    const int khalf = (lane >> 4) << 1;              // 0 or 2
    const float* arow = &Xs[m0 + row][khalf];

    // B fragment (4x16 f32): lanes 0-15 col N=lane @ K rows 0/1;
    //                        lanes 16-31 col N=lane-16 @ K rows 2/3.
    const int col = lane & 15;

    v8f acc0 = {}, acc1 = {}, acc2 = {}, acc3 = {};

    if constexpr (TR) {
        const float* b0 = &Wmat[(nbase +  0 + col) * D_IN + khalf];
        const float* b1 = &Wmat[(nbase + 16 + col) * D_IN + khalf];
        const float* b2 = &Wmat[(nbase + 32 + col) * D_IN + khalf];
        const float* b3 = &Wmat[(nbase + 48 + col) * D_IN + khalf];
        for (int k0 = 0; k0 < D_IN; k0 += 4) {
            const v2f a = *(const v2f*)(arow + k0);   // ds_load_b64
            v2f b;
            b = *(const v2f*)(b0 + k0);               // one global_load_b64 per WMMA
            acc0 = __builtin_amdgcn_wmma_f32_16x16x4_f32(false, a, false, b, (short)0, acc0, false, false);
            b = *(const v2f*)(b1 + k0);
            acc1 = __builtin_amdgcn_wmma_f32_16x16x4_f32(false, a, false, b, (short)0, acc1, false, false);
            b = *(const v2f*)(b2 + k0);
            acc2 = __builtin_amdgcn_wmma_f32_16x16x4_f32(false, a, false, b, (short)0, acc2, false, false);
            b = *(const v2f*)(b3 + k0);
            acc3 = __builtin_amdgcn_wmma_f32_16x16x4_f32(false, a, false, b, (short)0, acc3, false, false);
        }
    } else {
        for (int k0 = 0; k0 < D_IN; k0 += 4) {
            const v2f a = *(const v2f*)(arow + k0);
            const int kr = k0 + khalf;
            const float* bp = &Wmat[(long long)kr * D_OUT + nbase + col];
            v2f b;
            b.x = bp[0];  b.y = bp[D_OUT];
            acc0 = __builtin_amdgcn_wmma_f32_16x16x4_f32(false, a, false, b, (short)0, acc0, false, false);
            b.x = bp[16]; b.y = bp[D_OUT + 16];
            acc1 = __builtin_amdgcn_wmma_f32_16x16x4_f32(false, a, false, b, (short)0, acc1, false, false);
            b.x = bp[32]; b.y = bp[D_OUT + 32];
            acc2 = __builtin_amdgcn_wmma_f32_16x16x4_f32(false, a, false, b, (short)0, acc2, false, false);
            b.x = bp[48]; b.y = bp[D_OUT + 48];
            acc3 = __builtin_amdgcn_wmma_f32_16x16x4_f32(false, a, false, b, (short)0, acc3, false, false);
        }
    }

    // ---------- Store with ReLU ----------
    // C/D layout: VGPR r -> rows (m0 + r) [lanes 0-15] and (m0 + r + 8) [lanes 16-31].
    const int rext = (lane >> 4) << 3;               // 0 or 8
    v8f accs[4] = {acc0, acc1, acc2, acc3};
    #pragma unroll
    for (int t = 0; t < 4; ++t) {
        const int n0 = nbase + t * 16;
        #pragma unroll
        for (int r = 0; r < 8; ++r) {
            const int grow = base + m0 + r + rext;
            if (grow < N_NODES) {
                const float v = accs[t][r];
                out[(long long)grow * D_OUT + n0 + col] = v > 0.f ? v : 0.f;
            }
        }
    }
}

extern "C" void kernel_launch(void* const* d_in, const int* in_sizes, int n_in,
                              void* d_out, int out_size, void* d_ws, size_t ws_size,
                              hipStream_t stream) {
    const float*     feat  = (const float*)d_in[0];
    const long long* edges = (const long long*)d_in[1];   // jax int64 indices
    const float*     Kmat  = (const float*)d_in[2];
    float*           out   = (float*)d_out;

    const int grid = (N_NODES + M_TILE - 1) / M_TILE;     // 1563 blocks

    if (ws_size >= (size_t)D_IN * D_OUT * sizeof(float)) {
        float* Wt = (float*)d_ws;
        transpose_w<<<D_OUT, D_IN, 0, stream>>>(Kmat, Wt);
        sage_fused_wmma<true><<<grid, 256, 0, stream>>>(feat, edges, Wt, out);
    } else {
        sage_fused_wmma<false><<<grid, 256, 0, stream>>>(feat, edges, Kmat, out);
    }
}